// SkeletonEstimator_56513179680820
// MI455X (gfx1250) — compile-verified
//
#include <hip/hip_runtime.h>

// ---------------- problem constants ----------------
#define B_   8
#define C_   19
#define H_   544
#define W_   960
#define THR  0.1f

// ---------------- tile geometry --------------------
// output tile per block: 32 x 32
// input  region : 52 x 68   (origin r0-10, c0-18), reflect-padded
// vblur  region : 48 x 64   (origin r0-8,  c0-16)  -> 3x4 grid of 16x16 WMMA tiles
// blur   region : 48 x 48   (origin r0-8,  c0-8)   -> 3x3 grid of 16x16 WMMA tiles
#define IN_H 52
#define IN_W 68
#define IN_S 68     // LDS row stride (floats)
#define V_H  48
#define V_W  64
#define V_S  65     // odd stride -> conflict-free column access
#define BL_H 48
#define BL_W 48
#define BL_S 49
#define NTHREADS 384

typedef float v2f __attribute__((ext_vector_type(2)));
typedef float v8f __attribute__((ext_vector_type(8)));

#define AS1 __attribute__((address_space(1)))
#define AS3 __attribute__((address_space(3)))

#if defined(__gfx1250__) && __has_builtin(__builtin_amdgcn_wmma_f32_16x16x4_f32)
#define HAVE_WMMA_F32X4 1
#else
#define HAVE_WMMA_F32X4 0
#endif

#if defined(__gfx1250__) && __has_builtin(__builtin_amdgcn_global_load_async_to_lds_b32)
#define HAVE_ASYNC_LDS 1
#else
#define HAVE_ASYNC_LDS 0
#endif

__device__ __forceinline__ int reflect_idx(int i, int n) {
    if (i < 0)  i = -i;            // jnp reflect: -1 -> 1, -2 -> 2
    if (i >= n) i = 2 * n - 2 - i; // n -> n-2, n+1 -> n-3
    return i;
}

// banded weight: w[i] for 0 <= i < 5 else 0 (select chain, stays in VGPRs)
__device__ __forceinline__ float bandw(int i, float w0, float w1, float w2,
                                       float w3, float w4) {
    float r = 0.0f;
    r = (i == 0) ? w0 : r;
    r = (i == 1) ? w1 : r;
    r = (i == 2) ? w2 : r;
    r = (i == 3) ? w3 : r;
    r = (i == 4) ? w4 : r;
    return r;
}

__global__ __launch_bounds__(NTHREADS)
void fused_blur_nms_kernel(const float* __restrict__ heat,
                           const float* __restrict__ k1d,
                           float* __restrict__ peaks_map,
                           float* __restrict__ maxes,
                           float* __restrict__ blurred_out)
{
    __shared__ float sIn[IN_H * IN_S];
    __shared__ float sV [V_H  * V_S ];
    __shared__ float sB [BL_H * BL_S];

    const int t    = threadIdx.x;
    const int wave = t >> 5;        // wave32
    const int lane = t & 31;
    const int r0   = blockIdx.y * 32;
    const int c0   = blockIdx.x * 32;
    const int img  = blockIdx.z;    // flattened (b, c)
    const float* __restrict__ src = heat + (size_t)img * (H_ * W_);

    const float w0 = k1d[0], w1 = k1d[1], w2 = k1d[2], w3 = k1d[3], w4 = k1d[4];

    // ---- stage 0: stage reflect-padded input region into LDS ----
    // CDNA5 async gather: memory -> LDS directly, tracked by ASYNCcnt.
    for (int idx = t; idx < IN_H * IN_W; idx += NTHREADS) {
        int lr = idx / IN_W;
        int lc = idx - lr * IN_W;
        int gr = reflect_idx(r0 - 10 + lr, H_);
        int gc = reflect_idx(c0 - 18 + lc, W_);
        const float* gp = src + (size_t)gr * W_ + gc;
        float* lp = &sIn[lr * IN_S + lc];
#if HAVE_ASYNC_LDS
        __builtin_amdgcn_global_load_async_to_lds_b32(
            (AS1 int*)(uintptr_t)gp, (AS3 int*)lp, 0, 0);
#else
        *lp = *gp;
#endif
    }
#if HAVE_ASYNC_LDS
#if __has_builtin(__builtin_amdgcn_s_wait_asynccnt)
    __builtin_amdgcn_s_wait_asynccnt(0);
#else
    asm volatile("s_wait_asynccnt 0" ::: "memory");
#endif
#endif
    __syncthreads();

    const int half = lane >> 4;      // 0: lanes 0-15, 1: lanes 16-31
    const int mn   = lane & 15;      // M (for A/D) or N (for B/C) index
    const int k0   = half * 2;       // K base held by this lane group

    // Per-lane banded weight fragments, shared by vertical-A and horizontal-B:
    // chunk c covers K = 4c..4c+3; this lane's VGPR0 holds K=4c+k0, VGPR1 K=4c+k0+1.
    float wA0[5], wA1[5];
    #pragma unroll
    for (int c = 0; c < 5; ++c) {
        int j0 = 4 * c + k0;
        wA0[c] = bandw(j0 - mn,     w0, w1, w2, w3, w4);
        wA1[c] = bandw(j0 + 1 - mn, w0, w1, w2, w3, w4);
    }

    // ---- stage 1: vertical blur  Vblur(16x16) = Wv(16x20) * In(20x16) ----
    {
        const int ti = wave >> 2;   // 0..2  (row tile)
        const int tj = wave & 3;    // 0..3  (col tile)
#if HAVE_WMMA_F32X4
        v8f acc = {0.f, 0.f, 0.f, 0.f, 0.f, 0.f, 0.f, 0.f};
        #pragma unroll
        for (int c = 0; c < 5; ++c) {
            // B fragment: In rows (window) 4c+K, cols mn
            int br = 16 * ti + 4 * c + k0;
            int bc = 2 + 16 * tj + mn;
            v2f A  = { wA0[c], wA1[c] };
            v2f Bm = { sIn[br * IN_S + bc], sIn[(br + 1) * IN_S + bc] };
            acc = __builtin_amdgcn_wmma_f32_16x16x4_f32(
                      false, A, false, Bm, (short)0, acc, false, false);
        }
        #pragma unroll
        for (int v = 0; v < 8; ++v)
            sV[(16 * ti + v + 8 * half) * V_S + 16 * tj + mn] = acc[v];
#else
        // fallback: direct 5-tap vertical conv producing the same D layout
        #pragma unroll
        for (int v = 0; v < 8; ++v) {
            int m  = v + 8 * half;
            int bc = 2 + 16 * tj + mn;
            int rb = 16 * ti + m;
            float s = w0 * sIn[(rb + 0) * IN_S + bc];
            s = fmaf(w1, sIn[(rb + 1) * IN_S + bc], s);
            s = fmaf(w2, sIn[(rb + 2) * IN_S + bc], s);
            s = fmaf(w3, sIn[(rb + 3) * IN_S + bc], s);
            s = fmaf(w4, sIn[(rb + 4) * IN_S + bc], s);
            sV[(16 * ti + m) * V_S + 16 * tj + mn] = s;
        }
#endif
    }
    __syncthreads();

    // ---- stage 2: horizontal blur  Blur(16x16) = Vblur(16x20) * Wh(20x16) ----
    if (wave < 9) {
        const int si = wave / 3;
        const int sj = wave - si * 3;
#if HAVE_WMMA_F32X4
        v8f acc = {0.f, 0.f, 0.f, 0.f, 0.f, 0.f, 0.f, 0.f};
        #pragma unroll
        for (int c = 0; c < 5; ++c) {
            // A fragment: Vblur row mn, window cols 4c+K  (window starts at 6+16*sj)
            int ar = 16 * si + mn;
            int ac = 6 + 16 * sj + 4 * c + k0;
            v2f A  = { sV[ar * V_S + ac], sV[ar * V_S + ac + 1] };
            v2f Bm = { wA0[c], wA1[c] };   // Wh[j][n] = w[j-n], same band as Wv
            acc = __builtin_amdgcn_wmma_f32_16x16x4_f32(
                      false, A, false, Bm, (short)0, acc, false, false);
        }
        #pragma unroll
        for (int v = 0; v < 8; ++v)
            sB[(16 * si + v + 8 * half) * BL_S + 16 * sj + mn] = acc[v];
#else
        #pragma unroll
        for (int v = 0; v < 8; ++v) {
            int m  = v + 8 * half;
            int ar = 16 * si + m;
            int cb = 6 + 16 * sj + mn;   // window col for n=mn is (8+16sj+mn)-2
            float s = w0 * sV[ar * V_S + cb + 0];
            s = fmaf(w1, sV[ar * V_S + cb + 1], s);
            s = fmaf(w2, sV[ar * V_S + cb + 2], s);
            s = fmaf(w3, sV[ar * V_S + cb + 3], s);
            s = fmaf(w4, sV[ar * V_S + cb + 4], s);
            sB[(16 * si + m) * BL_S + 16 * sj + mn] = s;
        }
#endif
    }
    __syncthreads();

    // ---- stage 3: 3x3 max-pool + NMS + threshold, float4 output streams ----
    // 32 rows x 8 col-groups = 256 work items; thread t < 256 handles one group of 4.
    if (t < 256) {
        const int rr = t >> 3;          // 0..31
        const int cg = (t & 7) * 4;     // 0,4,...,28
        const int br = rr + 8, bc = cg + 8;   // into 48x48 blurred region

        // 3x6 neighborhood -> vertical maxes, then horizontal triples
        float colmax[6], ctr[4];
        #pragma unroll
        for (int dc = 0; dc < 6; ++dc) {
            float a = sB[(br - 1) * BL_S + (bc - 1 + dc)];
            float b = sB[(br    ) * BL_S + (bc - 1 + dc)];
            float c = sB[(br + 1) * BL_S + (bc - 1 + dc)];
            colmax[dc] = fmaxf(fmaxf(a, b), c);
            if (dc >= 1 && dc <= 4) ctr[dc - 1] = b;
        }
        float4 pkv, mxv, ctv;
        float* pk = (float*)&pkv;
        float* mx = (float*)&mxv;
        float* ct = (float*)&ctv;
        #pragma unroll
        for (int j = 0; j < 4; ++j) {
            float m = fmaxf(fmaxf(colmax[j], colmax[j + 1]), colmax[j + 2]);
            float c = ctr[j];
            mx[j] = m;
            ct[j] = c;
            pk[j] = (c == m && c > THR) ? c : 0.0f;
        }
        const size_t o = (size_t)img * (H_ * W_) + (size_t)(r0 + rr) * W_ + (c0 + cg);
        *(float4*)(peaks_map   + o) = pkv;   // 16B-aligned: W and col group % 4 == 0
        *(float4*)(maxes       + o) = mxv;
        *(float4*)(blurred_out + o) = ctv;
    }
}

extern "C" void kernel_launch(void* const* d_in, const int* in_sizes, int n_in,
                              void* d_out, int out_size, void* d_ws, size_t ws_size,
                              hipStream_t stream) {
    (void)in_sizes; (void)n_in; (void)out_size; (void)d_ws; (void)ws_size;
    const float* heat = (const float*)d_in[0];
    const float* k1d  = (const float*)d_in[1];
    float* out = (float*)d_out;
    const size_t N = (size_t)B_ * C_ * H_ * W_;
    dim3 grid(W_ / 32, H_ / 32, B_ * C_);   // (30, 17, 152)
    fused_blur_nms_kernel<<<grid, NTHREADS, 0, stream>>>(
        heat, k1d, out /*peaks_map*/, out + N /*maxes*/, out + 2 * N /*blurred*/);
}